// TransformerBlock_53317724012597
// MI455X (gfx1250) — compile-verified
//
#include <hip/hip_runtime.h>
#include <hip/hip_bf16.h>
#include <math.h>

// ---------------------------------------------------------------------------
// Transformer block (pre-LN GPT) for gfx1250, fp32 via V_WMMA_F32_16X16X4_F32
// B=2, S=2048, D=2048, H=16, HD=128
// ---------------------------------------------------------------------------

typedef float v2f __attribute__((ext_vector_type(2)));
typedef float v8f __attribute__((ext_vector_type(8)));

#define BATCH 2
#define SEQ   2048
#define DMODEL 2048
#define NHEAD 16
#define HDIM  128

#define EPI_PLAIN 0
#define EPI_QKV   1
#define EPI_RESID 2
#define EPI_GELU  3

// ---------------------------------------------------------------------------
// LayerNorm: one row (2048 floats) per 256-thread block (8 wave32s)
// ---------------------------------------------------------------------------
__global__ __launch_bounds__(256)
void ln_f32_kernel(const float* __restrict__ x, const float* __restrict__ w,
                   const float* __restrict__ bvec, float* __restrict__ out)
{
    __shared__ float red1[8];
    __shared__ float red2[8];
    const int tid  = threadIdx.x;
    const int lane = tid & 31;
    const int wv   = tid >> 5;
    const size_t row = blockIdx.x;
    const float* xr = x + row * (size_t)DMODEL;

    float vals[8];
    float s = 0.f;
#pragma unroll
    for (int i = 0; i < 8; ++i) { vals[i] = xr[tid + 256 * i]; s += vals[i]; }
#pragma unroll
    for (int off = 16; off >= 1; off >>= 1) s += __shfl_xor(s, off, 32);
    if (lane == 0) red1[wv] = s;
    __syncthreads();
    float tot = 0.f;
#pragma unroll
    for (int i = 0; i < 8; ++i) tot += red1[i];
    const float mean = tot * (1.0f / (float)DMODEL);

    float s2 = 0.f;
#pragma unroll
    for (int i = 0; i < 8; ++i) { float d = vals[i] - mean; s2 += d * d; }
#pragma unroll
    for (int off = 16; off >= 1; off >>= 1) s2 += __shfl_xor(s2, off, 32);
    if (lane == 0) red2[wv] = s2;
    __syncthreads();
    float tot2 = 0.f;
#pragma unroll
    for (int i = 0; i < 8; ++i) tot2 += red2[i];
    const float rstd = rsqrtf(tot2 * (1.0f / (float)DMODEL) + 1e-5f);

    float* orow = out + row * (size_t)DMODEL;
#pragma unroll
    for (int i = 0; i < 8; ++i) {
        const int c = tid + 256 * i;
        orow[c] = w[c] * (vals[i] - mean) * rstd + bvec[c];
    }
}

// ---------------------------------------------------------------------------
// fp32 WMMA GEMM: C[M,N] = A[M,K] @ Bw[K,N] + bias, with fused epilogues.
// Block = 256 threads = 8 waves; wave tile 32x32 (2x2 WMMA tiles);
// block tile 64 rows x 128 cols. All dims divide tiles (M=4096; N,K in
// {2048,6144,8192}).
// f32 A-frag layout (16x4): lane m=lane&15 holds A[m][2*ksel + {0,1}],
// ksel = lane>>4. B-frag (4x16) mirrored on columns. C/D: vgpr r holds
// row (r + 8*ksel), col = lane&15.
// ---------------------------------------------------------------------------
__global__ __launch_bounds__(256)
void gemm_f32_wmma_kernel(const float* __restrict__ A, const float* __restrict__ Bw,
                          const float* __restrict__ bias, const float* __restrict__ resid,
                          float* __restrict__ out0, float* __restrict__ out1,
                          float* __restrict__ out2,
                          int M, int N, int K, int mode)
{
    const int lane = threadIdx.x & 31;
    const int wave = threadIdx.x >> 5;
    const int rowBase = blockIdx.y * 64 + (wave >> 2) * 32;
    const int colBase = blockIdx.x * 128 + (wave & 3) * 32;
    const int m16  = lane & 15;
    const int ksel = lane >> 4;

    v8f acc00 = {0.f,0.f,0.f,0.f,0.f,0.f,0.f,0.f};
    v8f acc01 = acc00, acc10 = acc00, acc11 = acc00;

    const float* Arow0 = A + (size_t)(rowBase + m16) * K;
    const float* Arow1 = A + (size_t)(rowBase + 16 + m16) * K;
    const float* Bcol  = Bw + colBase + m16;

    for (int k0 = 0; k0 < K; k0 += 4) {
        const int ka = k0 + 2 * ksel;
        v2f a0 = *(const v2f*)(Arow0 + ka);
        v2f a1 = *(const v2f*)(Arow1 + ka);
        const float* bp = Bcol + (size_t)ka * N;
        v2f b0, b1;
        b0.x = bp[0];       b0.y = bp[N];
        b1.x = bp[16];      b1.y = bp[N + 16];
        acc00 = __builtin_amdgcn_wmma_f32_16x16x4_f32(false, a0, false, b0, (short)0, acc00, false, false);
        acc01 = __builtin_amdgcn_wmma_f32_16x16x4_f32(false, a0, false, b1, (short)0, acc01, false, false);
        acc10 = __builtin_amdgcn_wmma_f32_16x16x4_f32(false, a1, false, b0, (short)0, acc10, false, false);
        acc11 = __builtin_amdgcn_wmma_f32_16x16x4_f32(false, a1, false, b1, (short)0, acc11, false, false);
    }

#pragma unroll
    for (int t = 0; t < 4; ++t) {
        v8f acc = (t == 0) ? acc00 : (t == 1) ? acc01 : (t == 2) ? acc10 : acc11;
        const int ti = t >> 1, tj = t & 1;
#pragma unroll
        for (int r = 0; r < 8; ++r) {
            const int row = rowBase + ti * 16 + r + 8 * ksel;
            const int col = colBase + tj * 16 + m16;
            float v = acc[r] + bias[col];
            if (mode == EPI_QKV) {
                // col in [0,3D): section q/k/v, scatter to [B,H,S,HD]
                const int sec  = col >> 11;          // / DMODEL
                const int head = (col & 2047) >> 7;  // / HDIM
                const int hd   = col & 127;
                const int bb   = row >> 11;          // / SEQ
                const int ss   = row & 2047;
                const size_t dst = (((size_t)bb * NHEAD + head) * SEQ + ss) * HDIM + hd;
                float* o = (sec == 0) ? out0 : (sec == 1) ? out1 : out2;
                o[dst] = v;
            } else {
                const size_t idx = (size_t)row * N + col;
                if (mode == EPI_RESID)      v += resid[idx];
                else if (mode == EPI_GELU)  v = v / (1.0f + expf(-1.702f * v)); // quick_gelu
                out0[idx] = v;
            }
        }
    }
}

// ---------------------------------------------------------------------------
// Causal flash attention, fp32 WMMA. One wave per 16-query tile per (b,h).
// Q/K/V tiles (16x128) staged in padded LDS (stride 132 kills bank conflicts);
// scores 16x16 per key tile via 32 f32-WMMAs (K-dim=128); online softmax with
// 16-lane shfl_xor row reductions (C layout: vgpr r -> row r+8*(lane>>4));
// P pushed through LDS to re-form A-frags; PV via 8 col-tiles x 4 k-steps.
// ---------------------------------------------------------------------------
__global__ __launch_bounds__(32)
void attn_f32_wmma_kernel(const float* __restrict__ Q, const float* __restrict__ Kt,
                          const float* __restrict__ Vt, float* __restrict__ Y)
{
    __shared__ __attribute__((aligned(16))) float qs[16 * 132];
    __shared__ __attribute__((aligned(16))) float ks[16 * 132];
    __shared__ __attribute__((aligned(16))) float vs[16 * 132];
    __shared__ float ps[16 * 20];

    const int lane = threadIdx.x;
    const int qt = blockIdx.x;   // query tile (0..127)
    const int h  = blockIdx.y;
    const int b  = blockIdx.z;
    const int m16  = lane & 15;
    const int ksel = lane >> 4;

    const size_t headOff = ((size_t)b * NHEAD + h) * SEQ * HDIM;
    const float* Qb = Q + headOff + (size_t)qt * 16 * HDIM;

    // stage Q tile
    for (int i = lane; i < 512; i += 32) {
        const int r = i >> 5, c4 = (i & 31) * 4;
        *(float4*)(qs + r * 132 + c4) = *(const float4*)(Qb + r * HDIM + c4);
    }

    const float scale = 0.08838834764831845f; // 1/sqrt(128)
    float mrun[8], lrun[8];
    v8f zero8 = {0.f,0.f,0.f,0.f,0.f,0.f,0.f,0.f};
    v8f oacc[8];
#pragma unroll
    for (int r = 0; r < 8; ++r) { mrun[r] = -__builtin_inff(); lrun[r] = 0.f; }
#pragma unroll
    for (int ct = 0; ct < 8; ++ct) oacc[ct] = zero8;

    for (int kt = 0; kt <= qt; ++kt) {
        __syncthreads();
        const float* Kb = Kt + headOff + (size_t)kt * 16 * HDIM;
        const float* Vb = Vt + headOff + (size_t)kt * 16 * HDIM;
        for (int i = lane; i < 512; i += 32) {
            const int r = i >> 5, c4 = (i & 31) * 4;
            *(float4*)(ks + r * 132 + c4) = *(const float4*)(Kb + r * HDIM + c4);
            *(float4*)(vs + r * 132 + c4) = *(const float4*)(Vb + r * HDIM + c4);
        }
        __syncthreads();

        // scores = Q @ K^T : A=Q[16,128], B[k][n]=K[n][k] -> same LDS pattern
        v8f sacc = zero8;
        for (int k0 = 0; k0 < HDIM; k0 += 4) {
            const int kk = k0 + 2 * ksel;
            v2f a  = *(const v2f*)(qs + m16 * 132 + kk);
            v2f bb = *(const v2f*)(ks + m16 * 132 + kk);
            sacc = __builtin_amdgcn_wmma_f32_16x16x4_f32(false, a, false, bb, (short)0, sacc, false, false);
        }

        float pvals[8];
#pragma unroll
        for (int r = 0; r < 8; ++r) {
            float v = sacc[r] * scale;
            if (kt == qt) {
                const int mrow = r + 8 * ksel;        // query row in tile
                if (m16 > mrow) v = -__builtin_inff(); // causal mask on diagonal tile
            }
            pvals[r] = v;
        }

        // online softmax per row (rows live in 16-lane groups)
#pragma unroll
        for (int r = 0; r < 8; ++r) {
            float mx = pvals[r];
#pragma unroll
            for (int off = 1; off < 16; off <<= 1) mx = fmaxf(mx, __shfl_xor(mx, off, 32));
            const float mnew = fmaxf(mrun[r], mx);
            const float p = expf(pvals[r] - mnew);
            float sum = p;
#pragma unroll
            for (int off = 1; off < 16; off <<= 1) sum += __shfl_xor(sum, off, 32);
            const float alpha = expf(mrun[r] - mnew);
            lrun[r] = lrun[r] * alpha + sum;
            mrun[r] = mnew;
            pvals[r] = p;
#pragma unroll
            for (int ct = 0; ct < 8; ++ct) oacc[ct][r] *= alpha;
        }

        // P (C layout) -> LDS -> A-frag layout
        __syncthreads();
#pragma unroll
        for (int r = 0; r < 8; ++r) ps[(r + 8 * ksel) * 20 + m16] = pvals[r];
        __syncthreads();

        // out += P @ V
#pragma unroll
        for (int ct = 0; ct < 8; ++ct) {
#pragma unroll
            for (int k0 = 0; k0 < 16; k0 += 4) {
                const int kk = k0 + 2 * ksel;
                v2f a = *(const v2f*)(ps + m16 * 20 + kk);
                v2f bb;
                bb.x = vs[kk * 132 + ct * 16 + m16];
                bb.y = vs[(kk + 1) * 132 + ct * 16 + m16];
                oacc[ct] = __builtin_amdgcn_wmma_f32_16x16x4_f32(false, a, false, bb, (short)0, oacc[ct], false, false);
            }
        }
    }

    // normalize + store y[b, s, h*HD + d]
    float inv[8];
#pragma unroll
    for (int r = 0; r < 8; ++r) inv[r] = 1.0f / lrun[r];
#pragma unroll
    for (int ct = 0; ct < 8; ++ct) {
#pragma unroll
        for (int r = 0; r < 8; ++r) {
            const int mrow = r + 8 * ksel;
            const int srow = qt * 16 + mrow;
            const int d = ct * 16 + m16;
            Y[((size_t)b * SEQ + srow) * DMODEL + h * HDIM + d] = oacc[ct][r] * inv[r];
        }
    }
}

// ---------------------------------------------------------------------------
// Host-side orchestration
// ---------------------------------------------------------------------------
extern "C" void kernel_launch(void* const* d_in, const int* in_sizes, int n_in,
                              void* d_out, int out_size, void* d_ws, size_t ws_size,
                              hipStream_t stream)
{
    const float* x    = (const float*)d_in[0];
    const float* Wqkv = (const float*)d_in[1];
    const float* bqkv = (const float*)d_in[2];
    const float* Wo   = (const float*)d_in[3];
    const float* bo   = (const float*)d_in[4];
    const float* W1   = (const float*)d_in[5];
    const float* b1   = (const float*)d_in[6];
    const float* W2   = (const float*)d_in[7];
    const float* b2   = (const float*)d_in[8];
    const float* ln1w = (const float*)d_in[9];
    const float* ln1b = (const float*)d_in[10];
    const float* ln2w = (const float*)d_in[11];
    const float* ln2b = (const float*)d_in[12];

    const size_t TOK   = (size_t)BATCH * SEQ;          // 4096
    const size_t TOKD  = TOK * DMODEL;                 // 8,388,608

    // Workspace layout (floats), with lifetime-based reuse (~201 MB total):
    //   [0 .. TOKD)            h (ln1 out), later h2 (ln2 out)
    //   [TOKD .. 2*TOKD)       q [B,H,S,HD]; later x1 = x + attn (q is dead)
    //   [2*TOKD .. 3*TOKD)     y (attn out, [B,S,D]); later mid starts here
    //   [2*TOKD .. 6*TOKD)     mid [4096, 8192] (y is dead by MLP1)
    float* ws  = (float*)d_ws;
    float* h   = ws;                 // and h2
    float* q   = ws + TOKD;
    float* x1  = ws + TOKD;          // reuse of q region after attention
    float* y   = ws + 2 * TOKD;
    float* mid = ws + 2 * TOKD;      // reuse of y region after out-proj

    // d_out = concat(x_out [B,S,D], k [B,H,S,HD], v [B,H,S,HD])
    float* outx = (float*)d_out;
    float* outk = (float*)d_out + TOKD;
    float* outv = (float*)d_out + 2 * TOKD;

    // 1) h = LN1(x)
    ln_f32_kernel<<<dim3((unsigned)TOK), 256, 0, stream>>>(x, ln1w, ln1b, h);

    // 2) qkv = h @ Wqkv + bqkv, scattered: q->ws, k/v->d_out ([B,H,S,HD])
    gemm_f32_wmma_kernel<<<dim3(3 * DMODEL / 128, TOK / 64), 256, 0, stream>>>(
        h, Wqkv, bqkv, nullptr, q, outk, outv,
        (int)TOK, 3 * DMODEL, DMODEL, EPI_QKV);

    // 3) y = causal_attention(q, k, v)  (k,v read back from d_out)
    attn_f32_wmma_kernel<<<dim3(SEQ / 16, NHEAD, BATCH), 32, 0, stream>>>(
        q, outk, outv, y);

    // 4) x1 = x + (y @ Wo + bo)
    gemm_f32_wmma_kernel<<<dim3(DMODEL / 128, TOK / 64), 256, 0, stream>>>(
        y, Wo, bo, x, x1, nullptr, nullptr,
        (int)TOK, DMODEL, DMODEL, EPI_RESID);

    // 5) h2 = LN2(x1)
    ln_f32_kernel<<<dim3((unsigned)TOK), 256, 0, stream>>>(x1, ln2w, ln2b, h);

    // 6) mid = quick_gelu(h2 @ W1 + b1)
    gemm_f32_wmma_kernel<<<dim3(4 * DMODEL / 128, TOK / 64), 256, 0, stream>>>(
        h, W1, b1, nullptr, mid, nullptr, nullptr,
        (int)TOK, 4 * DMODEL, DMODEL, EPI_GELU);

    // 7) x_out = x1 + (mid @ W2 + b2)
    gemm_f32_wmma_kernel<<<dim3(DMODEL / 128, TOK / 64), 256, 0, stream>>>(
        mid, W2, b2, x1, outx, nullptr, nullptr,
        (int)TOK, DMODEL, 4 * DMODEL, EPI_RESID);
}